// ReadoutLayer_1151051235753
// MI455X (gfx1250) — compile-verified
//
#include <hip/hip_runtime.h>

#define D_DIM 128
#define TILE_ROWS 16
#define TILES_PER_CHUNK 8
#define CHUNK_ROWS (TILE_ROWS * TILES_PER_CHUNK)   // 128 rows per wave
#define WAVES_PER_BLOCK 4
#define BLOCK_THREADS (WAVES_PER_BLOCK * 32)       // wave32
#define TILE_STRIDE 132                            // 128 + 4-float pad: conflict-free transpose reads

typedef __attribute__((ext_vector_type(2))) float v2f;
typedef __attribute__((ext_vector_type(8))) float v8f;

__global__ void zero_ws_kernel(float* __restrict__ ws, int total) {
    int i = blockIdx.x * blockDim.x + threadIdx.x;
    if (i < total) ws[i] = 0.0f;
}

__device__ __forceinline__ void flush_seg(float* __restrict__ wssum,
                                          float* __restrict__ wscnt,
                                          int segid, const float4& acc, float cnt, int lane) {
    if (segid < 0) return;
    float* dst = wssum + (size_t)segid * D_DIM + lane * 4;
    atomicAdd(dst + 0, acc.x);
    atomicAdd(dst + 1, acc.y);
    atomicAdd(dst + 2, acc.z);
    atomicAdd(dst + 3, acc.w);
    if (lane == 0) atomicAdd(wscnt + segid, cnt);
}

__global__ void __launch_bounds__(BLOCK_THREADS)
readout_main_kernel(const float* __restrict__ h, const int* __restrict__ seg,
                    const float* __restrict__ att_w, const float* __restrict__ att_b,
                    float* __restrict__ wssum, float* __restrict__ wscnt, int n) {
    __shared__ float wbuf[D_DIM];
    __shared__ float tile[WAVES_PER_BLOCK][TILE_ROWS * TILE_STRIDE];

    if (threadIdx.x < D_DIM) wbuf[threadIdx.x] = att_w[threadIdx.x];
    __syncthreads();   // only barrier in the kernel: w staging

    const int wave = threadIdx.x >> 5;
    const int lane = threadIdx.x & 31;
    const int nchunks = (n + CHUNK_ROWS - 1) / CHUNK_ROWS;
    const int chunk = blockIdx.x * WAVES_PER_BLOCK + wave;
    if (chunk >= nchunks) return;

    const int row_begin = chunk * CHUNK_ROWS;
    const int row_end   = min(n, row_begin + CHUNK_ROWS);
    const int full_tiles = (row_end - row_begin) / TILE_ROWS;

    float* tw = tile[wave];               // wave-private tile: no cross-wave sync needed
    const float bconst = att_b[0];

    const int rowA  = lane & 15;          // A-matrix: lane selects M row
    const int khalf = lane >> 4;          // lanes 16-31 hold K=2,3 (float2 offset +2)

    float4 acc = make_float4(0.f, 0.f, 0.f, 0.f);
    float  cnt = 0.f;
    int    cur = -1;

    for (int t = 0; t < full_tiles; ++t) {
        const int row0 = row_begin + t * TILE_ROWS;

        // Coalesced read of 16 rows: one 512B row per wave per b128 load. Read HBM once.
        float4 hrow[TILE_ROWS];
        #pragma unroll
        for (int r = 0; r < TILE_ROWS; ++r)
            hrow[r] = *(const float4*)(h + (size_t)(row0 + r) * D_DIM + lane * 4);

        // Stage into wave-private LDS for the WMMA A-fragment transpose.
        #pragma unroll
        for (int r = 0; r < TILE_ROWS; ++r)
            *(float4*)&tw[r * TILE_STRIDE + lane * 4] = hrow[r];

        // 16 dot products via 32 accumulating f32 WMMAs:
        // D[m][n] += sum_k h[m][4kc+k] * w[4kc+k]  (B column-replicated w chunk)
        v8f c = {0.f, 0.f, 0.f, 0.f, 0.f, 0.f, 0.f, 0.f};
        #pragma unroll
        for (int kc = 0; kc < 32; ++kc) {
            v2f a  = *(const v2f*)&tw[rowA * TILE_STRIDE + kc * 4 + 2 * khalf];
            v2f bf = *(const v2f*)&wbuf[kc * 4 + 2 * khalf];
            c = __builtin_amdgcn_wmma_f32_16x16x4_f32(false, a, false, bf,
                                                      (short)0, c, false, false);
        }

        // att = exp(leaky_relu(dot + b)); c[v]: lanes 0-15 hold rows v, lanes 16-31 rows v+8
        float att8[8];
        #pragma unroll
        for (int v = 0; v < 8; ++v) {
            float s = c[v] + bconst;
            s = (s >= 0.f) ? s : 0.01f * s;
            att8[v] = __expf(s);
        }

        // Segment-run accumulation in registers; seg is sorted -> wave-uniform flush branch.
        #pragma unroll
        for (int r = 0; r < TILE_ROWS; ++r) {
            const float attr = __shfl(att8[r & 7], (r >> 3) << 4, 32);  // v_readlane broadcast
            const int sr = seg[row0 + r];
            if (sr != cur) {
                flush_seg(wssum, wscnt, cur, acc, cnt, lane);
                cur = sr;
                acc = make_float4(0.f, 0.f, 0.f, 0.f);
                cnt = 0.f;
            }
            acc.x = fmaf(attr, hrow[r].x, acc.x);
            acc.y = fmaf(attr, hrow[r].y, acc.y);
            acc.z = fmaf(attr, hrow[r].z, acc.z);
            acc.w = fmaf(attr, hrow[r].w, acc.w);
            cnt += 1.f;
        }
    }

    // Remainder rows (< 16, only the last chunk): scalar shuffle-reduce path.
    const float4 wl = *(const float4*)&wbuf[lane * 4];
    for (int row = row_begin + full_tiles * TILE_ROWS; row < row_end; ++row) {
        float4 hv = *(const float4*)(h + (size_t)row * D_DIM + lane * 4);
        float part = hv.x * wl.x + hv.y * wl.y + hv.z * wl.z + hv.w * wl.w;
        #pragma unroll
        for (int off = 16; off > 0; off >>= 1) part += __shfl_xor(part, off, 32);
        float s = part + bconst;
        s = (s >= 0.f) ? s : 0.01f * s;
        const float attr = __expf(s);
        const int sr = seg[row];
        if (sr != cur) {
            flush_seg(wssum, wscnt, cur, acc, cnt, lane);
            cur = sr;
            acc = make_float4(0.f, 0.f, 0.f, 0.f);
            cnt = 0.f;
        }
        acc.x = fmaf(attr, hv.x, acc.x);
        acc.y = fmaf(attr, hv.y, acc.y);
        acc.z = fmaf(attr, hv.z, acc.z);
        acc.w = fmaf(attr, hv.w, acc.w);
        cnt += 1.f;
    }

    flush_seg(wssum, wscnt, cur, acc, cnt, lane);
}

__global__ void finalize_kernel(const float* __restrict__ sums,
                                const float* __restrict__ cnts,
                                float* __restrict__ out, int total) {
    int i = blockIdx.x * blockDim.x + threadIdx.x;
    if (i < total) {
        float c = cnts[i >> 7];          // i / 128
        c = (c < 1.f) ? 1.f : c;
        out[i] = sums[i] / c;
    }
}

extern "C" void kernel_launch(void* const* d_in, const int* in_sizes, int n_in,
                              void* d_out, int out_size, void* d_ws, size_t ws_size,
                              hipStream_t stream) {
    const float* h     = (const float*)d_in[0];
    const int*   seg   = (const int*)d_in[1];
    const float* att_w = (const float*)d_in[2];
    const float* att_b = (const float*)d_in[3];
    float* out = (float*)d_out;

    const int n = in_sizes[1];               // N nodes (seg length)
    const int G = out_size / D_DIM;          // segments

    float* wssum = (float*)d_ws;                         // G*128 floats
    float* wscnt = wssum + (size_t)G * D_DIM;            // G floats

    const int ztotal = G * (D_DIM + 1);
    zero_ws_kernel<<<(ztotal + 255) / 256, 256, 0, stream>>>(wssum, ztotal);

    const int nchunks = (n + CHUNK_ROWS - 1) / CHUNK_ROWS;
    const int blocks  = (nchunks + WAVES_PER_BLOCK - 1) / WAVES_PER_BLOCK;
    readout_main_kernel<<<blocks, BLOCK_THREADS, 0, stream>>>(h, seg, att_w, att_b,
                                                              wssum, wscnt, n);

    finalize_kernel<<<(out_size + 255) / 256, 256, 0, stream>>>(wssum, wscnt, out, out_size);
}